// DSBatchNorm_9320079033284
// MI455X (gfx1250) — compile-verified
//
#include <hip/hip_runtime.h>

typedef float v2f __attribute__((ext_vector_type(2)));
typedef float v8f __attribute__((ext_vector_type(8)));

#define N_ROWS   262144
#define NF       512
#define NDOM     8
#define FGROUPS  32                      // 512 / 16 features per WMMA tile
#define CHUNKS   128
#define ROWS_PER_CHUNK (N_ROWS / CHUNKS) // 2048
#define STEPS    (ROWS_PER_CHUNK / 4)    // 512 WMMA steps (K=4 rows each)
#define EPSV     1e-5f

// ---------------------------------------------------------------------------
// Phase 1: segmented sum / sumsq via V_WMMA_F32_16X16X4_F32 one-hot matmul.
// One wave per (chunk, feature-group-of-16). A = one-hot(domain of 4 rows),
// B = 4 rows x 16 features of x (or x^2). C accumulates [16 dom x 16 feat].
// ---------------------------------------------------------------------------
__global__ __launch_bounds__(32) void p1_segreduce(
    const float* __restrict__ x, const int* __restrict__ y,
    float* __restrict__ psum, float* __restrict__ psq, float* __restrict__ pcnt) {
  const int chunk = blockIdx.x / FGROUPS;
  const int fg    = blockIdx.x % FGROUPS;
  const int lane  = threadIdx.x;
  const int m     = lane & 15;   // A row (domain) / B col (feature)
  const int hi    = lane >> 4;   // 0: K=0,1   1: K=2,3
  const int f0    = fg * 16;
  const int rowbase = chunk * ROWS_PER_CHUNK;

  v8f csum = {};
  v8f csq  = {};
  float vcnt = 0.0f;

  // lane reads B[k = 2*hi + v][n = m] = x[row + 2*hi + v][f0 + m]
  const float* px = x + (size_t)(rowbase + 2 * hi) * NF + f0 + m;

  for (int s = 0; s < STEPS; ++s) {
    const int rb = rowbase + s * 4;
    const int4 yv = *(const int4*)(y + rb);          // uniform -> scalar load
    const int ya = hi ? yv.z : yv.x;                 // y[rb + 2*hi]
    const int yb = hi ? yv.w : yv.y;                 // y[rb + 2*hi + 1]

    v2f a;                                           // one-hot A tile
    a.x = (ya == m) ? 1.0f : 0.0f;
    a.y = (yb == m) ? 1.0f : 0.0f;

    v2f b;                                           // x tile
    b.x = px[0];
    b.y = px[NF];
    px += 4 * NF;

    v2f b2;                                          // x^2 tile
    b2.x = b.x * b.x;
    b2.y = b.y * b.y;

    csum = __builtin_amdgcn_wmma_f32_16x16x4_f32(false, a, false, b,  (short)0, csum, false, false);
    csq  = __builtin_amdgcn_wmma_f32_16x16x4_f32(false, a, false, b2, (short)0, csq,  false, false);
    vcnt += a.x + a.y;                               // free count histogram
  }

  // C layout: lane l, vgpr v -> domain = v + 8*(l>=16), feature = f0 + (l&15).
  // Domains 8..15 (lanes 16..31) are identically zero -> store lanes 0..15 only.
  if (lane < 16) {
    const size_t base = ((size_t)chunk * NDOM) * NF + f0 + m;
#pragma unroll
    for (int v = 0; v < NDOM; ++v) {
      psum[base + (size_t)v * NF] = csum[v];
      psq [base + (size_t)v * NF] = csq[v];
    }
  }
  // count[d] = vcnt(lane d: K=0,1 part) + vcnt(lane d+16: K=2,3 part)
  const float tot = vcnt + __shfl_xor(vcnt, 16, 32);
  if (fg == 0 && lane < NDOM) pcnt[chunk * NDOM + lane] = tot;
}

// ---------------------------------------------------------------------------
// Phase 2: deterministic fixed-order reduce of chunk partials, then fold BN
// statistics + gamma/beta into per-(domain,feature) scale/shift tables.
// ---------------------------------------------------------------------------
__global__ __launch_bounds__(256) void p2_stats(
    const float* __restrict__ psum, const float* __restrict__ psq,
    const float* __restrict__ pcnt,
    const float* __restrict__ gamma, const float* __restrict__ beta,
    float* __restrict__ scale, float* __restrict__ shift) {
  const int t = blockIdx.x * blockDim.x + threadIdx.x;   // 0 .. 4095
  const int d = t >> 9;
  const int f = t & (NF - 1);

  float s = 0.0f, q = 0.0f, c = 0.0f;
  for (int ch = 0; ch < CHUNKS; ++ch) {
    const size_t o = ((size_t)ch * NDOM + d) * NF + f;
    s += psum[o];
    q += psq[o];
    c += pcnt[ch * NDOM + d];
  }

  const float safe = fmaxf(c, 1.0f);
  const float mean = s / safe;
  const float var  = q / safe - mean * mean;
  const bool  use_batch = (c > 1.0f);
  const float mean_e = use_batch ? mean : 0.0f;
  const float var_e  = use_batch ? var  : 1.0f;
  const float inv = rsqrtf(var_e + EPSV);

  const int idx = d * NF + f;
  float scl = gamma[idx] * inv;
  float shf = beta[idx] - mean_e * scl;
  if (!(c > 0.0f)) { scl = 0.0f; shf = 0.0f; }         // empty domain -> 0 out
  scale[idx] = scl;
  shift[idx] = shf;
}

// ---------------------------------------------------------------------------
// Phase 3: out[n,f] = x[n,f] * scale[y[n],f] + shift[y[n],f].
// 32 KB scale/shift tables staged in LDS; float4 streaming (b128 loads/stores).
// ---------------------------------------------------------------------------
__global__ __launch_bounds__(256) void p3_normalize(
    const float* __restrict__ x, const int* __restrict__ y,
    const float* __restrict__ scale, const float* __restrict__ shift,
    float* __restrict__ out) {
  __shared__ float4 sS[NDOM * 128];
  __shared__ float4 sH[NDOM * 128];
  const float4* s4 = (const float4*)scale;
  const float4* h4 = (const float4*)shift;
  for (int i = threadIdx.x; i < NDOM * 128; i += 256) {
    sS[i] = s4[i];
    sH[i] = h4[i];
  }
  __syncthreads();

  const int q   = threadIdx.x & 127;   // float4 column (128 * 4 = 512 feats)
  const int sub = threadIdx.x >> 7;    // two rows in flight per iteration
  const int rowbase = blockIdx.x * 128;
  const float4* x4 = (const float4*)x;
  float4* o4 = (float4*)out;

  for (int rl = sub; rl < 128; rl += 2) {
    const int r = rowbase + rl;
    const int d = y[r];
    const size_t off = (size_t)r * 128 + q;
    const float4 xv = x4[off];
    const float4 sc = sS[d * 128 + q];
    const float4 sh = sH[d * 128 + q];
    float4 ov;
    ov.x = fmaf(xv.x, sc.x, sh.x);
    ov.y = fmaf(xv.y, sc.y, sh.y);
    ov.z = fmaf(xv.z, sc.z, sh.z);
    ov.w = fmaf(xv.w, sc.w, sh.w);
    o4[off] = ov;
  }
}

// ---------------------------------------------------------------------------
extern "C" void kernel_launch(void* const* d_in, const int* in_sizes, int n_in,
                              void* d_out, int out_size, void* d_ws, size_t ws_size,
                              hipStream_t stream) {
  const float* x     = (const float*)d_in[0];
  const int*   y     = (const int*)d_in[1];
  const float* gamma = (const float*)d_in[2];
  const float* beta  = (const float*)d_in[3];
  float* out = (float*)d_out;

  // workspace layout (floats)
  float* psum  = (float*)d_ws;                                  // CHUNKS*8*512
  float* psq   = psum + (size_t)CHUNKS * NDOM * NF;             // CHUNKS*8*512
  float* pcnt  = psq  + (size_t)CHUNKS * NDOM * NF;             // CHUNKS*8
  float* scale = pcnt + (size_t)CHUNKS * NDOM;                  // 8*512
  float* shift = scale + (size_t)NDOM * NF;                     // 8*512

  p1_segreduce<<<CHUNKS * FGROUPS, 32, 0, stream>>>(x, y, psum, psq, pcnt);
  p2_stats<<<(NDOM * NF) / 256, 256, 0, stream>>>(psum, psq, pcnt, gamma, beta, scale, shift);
  p3_normalize<<<N_ROWS / 128, 256, 0, stream>>>(x, y, scale, shift, out);
}